// FixedRatioSelector_76982993814146
// MI455X (gfx1250) — compile-verified
//
#include <hip/hip_runtime.h>
#include <stdint.h>

// Problem constants (reference: B=4, S=2048, RATIO=0.5 -> k=512)
#define S_DIM    2048
#define K_SEL    512
#define NTHREADS 256
#define NWAVES   (NTHREADS / 32)      // 8 wave32s
#define EPT      (S_DIM / NTHREADS)   // 8 elements per thread

static_assert(EPT == 8, "layout assumes 8 elements per thread");

// Monotone map: descending float order == descending uint order.
__device__ __forceinline__ uint32_t mono_map(uint32_t b) {
  return b ^ ((b & 0x80000000u) ? 0xFFFFFFFFu : 0x80000000u);
}

// Lane shuffle via ds_bpermute (wave32; index in bytes, bits[6:2] used).
__device__ __forceinline__ uint32_t lane_read(uint32_t v, int srcLane) {
  return (uint32_t)__builtin_amdgcn_ds_bpermute((srcLane & 31) << 2, (int)v);
}

__global__ __launch_bounds__(NTHREADS)
void FixedRatioSelector_76982993814146_kernel(const float* __restrict__ scores,
                                              float* __restrict__ out_mask,
                                              float* __restrict__ out_k) {
  const int row  = blockIdx.x;          // row = b*S + q
  const int q    = row & (S_DIM - 1);
  const int t    = threadIdx.x;
  const int lane = t & 31;
  const int wave = t >> 5;

  float* __restrict__ orow = out_mask + (size_t)row * S_DIM;
  if (t == 0) out_k[row] = (float)K_SEL;   // k_values is constant 512

  // ---- Rows with q < k: all visible keys selected; no score read needed ----
  if (q < K_SEL) {
#pragma unroll
    for (int c = 0; c < 2; ++c) {
      const int j0 = t * EPT + c * 4;
      float4 v;
      v.x = (j0 + 0 <= q) ? 1.0f : 0.0f;
      v.y = (j0 + 1 <= q) ? 1.0f : 0.0f;
      v.z = (j0 + 2 <= q) ? 1.0f : 0.0f;
      v.w = (j0 + 3 <= q) ? 1.0f : 0.0f;
      *reinterpret_cast<float4*>(orow + j0) = v;   // global_store_b128
    }
    return;
  }

  // ---- Rows with q >= k: exact top-512 via 4-pass radix select ----
  __shared__ __align__(16) uint32_t sh_u[S_DIM];   // async staging (8 KB)
  __shared__ uint32_t sh_hist[256];                // radix histogram
  __shared__ uint32_t sh_wsum[NWAVES];             // per-wave scan partials
  __shared__ uint32_t sh_state[2];                 // {prefix, remaining k}

  const float* __restrict__ srow = scores + (size_t)row * S_DIM;

  // Async stage the whole row into LDS: 2 x b128 per lane (ASYNCcnt path).
  {
    uint32_t l0 = (uint32_t)(uintptr_t)(&sh_u[t * 4]);              // LDS byte addr
    uint32_t l1 = (uint32_t)(uintptr_t)(&sh_u[t * 4 + (S_DIM / 2)]);
    const float* g0 = srow + t * 4;
    const float* g1 = srow + t * 4 + (S_DIM / 2);
    asm volatile("global_load_async_to_lds_b128 %0, %1, off"
                 :: "v"(l0), "v"(g0) : "memory");
    asm volatile("global_load_async_to_lds_b128 %0, %1, off"
                 :: "v"(l1), "v"(g1) : "memory");
    asm volatile("s_wait_asynccnt 0" ::: "memory");
  }
  if (t == 0) { sh_state[0] = 0u; sh_state[1] = (uint32_t)K_SEL; }
  __syncthreads();

  // Pull this thread's contiguous chunk (two ds_load_b128), monotone-mapped;
  // invisible keys (j > q) -> 0 (below any real score incl. -inf's 0x007FFFFF).
  uint32_t uv[EPT];
  {
    const uint4* up = reinterpret_cast<const uint4*>(&sh_u[t * EPT]);
    uint4 a = up[0], b = up[1];
    uint32_t raw[EPT] = {a.x, a.y, a.z, a.w, b.x, b.y, b.z, b.w};
#pragma unroll
    for (int i = 0; i < EPT; ++i) {
      const int j = t * EPT + i;
      uv[i] = (j <= q) ? mono_map(raw[i]) : 0u;
    }
  }
  __syncthreads();

  // 4 radix passes, MSB first: find exact uint T of the k-th largest value.
  for (int shift = 24; shift >= 0; shift -= 8) {
    const uint32_t prefix = sh_state[0];
    const uint32_t remK   = sh_state[1];
    sh_hist[t] = 0u;
    __syncthreads();

    const uint32_t pmask = (shift == 24) ? 0u : (0xFFFFFFFFu << (shift + 8));
#pragma unroll
    for (int i = 0; i < EPT; ++i) {
      const uint32_t u = uv[i];
      if ((u & pmask) == prefix)
        atomicAdd(&sh_hist[(u >> shift) & 0xFFu], 1u);
    }
    __syncthreads();

    // Suffix cumulative over 256 bins (bin == t):
    // intra-wave inclusive suffix scan (ds_bpermute, no barriers) ...
    const uint32_t h = sh_hist[t];
    uint32_t v = h;
#pragma unroll
    for (int off = 1; off < 32; off <<= 1) {
      const uint32_t o = lane_read(v, lane + off);
      v += (lane + off < 32) ? o : 0u;
    }
    // ... then combine totals of higher waves (bins above this wave's 32).
    if (lane == 0) sh_wsum[wave] = v;   // wave total = lane0 suffix
    __syncthreads();
    uint32_t higher = 0;
#pragma unroll
    for (int w = 0; w < NWAVES; ++w) higher += (w > wave) ? sh_wsum[w] : 0u;

    const uint32_t incl = v + higher;   // count in bins >= t
    const uint32_t excl = incl - h;     // count in bins  > t
    if (excl < remK && incl >= remK) {  // unique winning bin
      sh_state[0] = prefix | ((uint32_t)t << shift);
      sh_state[1] = remK - excl;
    }
    __syncthreads();
  }

  const uint32_t T       = sh_state[0];  // exact k-th largest (u-space)
  const uint32_t numTies = sh_state[1];  // how many ==T elements to keep

  // Stable tie-break by index: forward prefix scan of per-chunk tie counts
  // (chunks are index-ordered by thread id, elements ordered within chunk).
  uint32_t cnt = 0;
#pragma unroll
  for (int i = 0; i < EPT; ++i) cnt += (uv[i] == T) ? 1u : 0u;
  uint32_t c = cnt;
#pragma unroll
  for (int off = 1; off < 32; off <<= 1) {
    const uint32_t o = lane_read(c, lane - off);
    c += (lane >= off) ? o : 0u;
  }
  if (lane == 31) sh_wsum[wave] = c;    // wave total = lane31 prefix
  __syncthreads();
  uint32_t lower = 0;
#pragma unroll
  for (int w = 0; w < NWAVES; ++w) lower += (w < wave) ? sh_wsum[w] : 0u;
  uint32_t tieRank = lower + c - cnt;   // index-ordered exclusive rank

  // Emit mask: u > T always selected; ==T selected iff among first numTies.
  float f[EPT];
#pragma unroll
  for (int i = 0; i < EPT; ++i) {
    const uint32_t u = uv[i];
    float sel;
    if (u > T)        sel = 1.0f;
    else if (u == T) { sel = (tieRank < numTies) ? 1.0f : 0.0f; ++tieRank; }
    else              sel = 0.0f;
    f[i] = sel;
  }
  float4 o0 = {f[0], f[1], f[2], f[3]};
  float4 o1 = {f[4], f[5], f[6], f[7]};
  float4* __restrict__ ov = reinterpret_cast<float4*>(orow + t * EPT);
  ov[0] = o0;                            // global_store_b128 x2
  ov[1] = o1;
}

extern "C" void kernel_launch(void* const* d_in, const int* in_sizes, int n_in,
                              void* d_out, int out_size, void* d_ws, size_t ws_size,
                              hipStream_t stream) {
  (void)n_in; (void)d_ws; (void)ws_size; (void)out_size;
  // d_in[0] = x [B,2048,1024] f32 (unused), d_in[1] = index_scores [B,2048,2048] f32
  const float* scores = (const float*)d_in[1];
  const int B = in_sizes[1] / (S_DIM * S_DIM);
  float* out_mask = (float*)d_out;                        // [B,S,S] as 0/1 floats
  float* out_k    = out_mask + (size_t)B * S_DIM * S_DIM; // [B,S] == 512
  dim3 grid(B * S_DIM), block(NTHREADS);
  hipLaunchKernelGGL(FixedRatioSelector_76982993814146_kernel, grid, block, 0, stream,
                     scores, out_mask, out_k);
}